// YoloLossModel_20547123544316
// MI455X (gfx1250) — compile-verified
//
#include <hip/hip_runtime.h>
#include <hip/hip_bf16.h>

// YOLO loss, CDNA5 (gfx1250, wave32).
// k1: box prep (corner form + areas + conf)
// k2a: row-segmented 6272x6272 cross-IoU max/argmax (LDS-tiled VALU), 8-way row split
// k2b: segment combine (first-max semantics) + coord losses, block partials
// k3: class/noobj SSE via V_WMMA_F32_16X16X4_F32 (B = ones summation trick), 32-bit index math
// k4: deterministic fixed-order combine of block partials -> scalar

#define N_CH 30
#define TILE 256
#define NSEG 8

typedef float v2f __attribute__((ext_vector_type(2)));
typedef float v8f __attribute__((ext_vector_type(8)));

__global__ void yolo_prep(const float* __restrict__ P, const float* __restrict__ T,
                          int NB, float4* __restrict__ BP, float4* __restrict__ BT,
                          float* __restrict__ ConfP, float* __restrict__ A1,
                          float* __restrict__ A2) {
  int j = blockIdx.x * blockDim.x + threadIdx.x;
  if (j >= NB) return;
  int k = j >> 1;        // obj-cell ordinal
  int b = j & 1;         // box slot within cell
  int rowbase = (k << 3) * N_CH;          // obj cells are every 8th cell
  int off  = b ? 5 : 0;
  int ccol = b ? 9 : 4;

  float px = P[rowbase + off + 0], py = P[rowbase + off + 1];
  float pw = P[rowbase + off + 2], ph = P[rowbase + off + 3];
  float tx = T[rowbase + off + 0], ty = T[rowbase + off + 1];
  float tw = T[rowbase + off + 2], th = T[rowbase + off + 3];

  // corner form, matching reference rounding: xy/S -/+ 0.5*wh
  float pcx = px / 7.0f, pcy = py / 7.0f;
  float tcx = tx / 7.0f, tcy = ty / 7.0f;
  float4 bp = make_float4(pcx - 0.5f * pw, pcy - 0.5f * ph,
                          pcx + 0.5f * pw, pcy + 0.5f * ph);
  float4 bt = make_float4(tcx - 0.5f * tw, tcy - 0.5f * th,
                          tcx + 0.5f * tw, tcy + 0.5f * th);
  BP[j] = bp;
  BT[j] = bt;
  // areas from corners (same fp path as reference)
  A1[j] = (bp.z - bp.x) * (bp.w - bp.y);
  A2[j] = (bt.z - bt.x) * (bt.w - bt.y);
  ConfP[j] = P[rowbase + ccol];
}

// Each (blockIdx.x column-tile, blockIdx.y segment) computes per-column max/argmax
// over rows [seg*segLen, min((seg+1)*segLen, NB)), ascending (first-max).
__global__ void yolo_iou_partial(const float4* __restrict__ BP, const float4* __restrict__ BT,
                                 const float* __restrict__ A1, const float* __restrict__ A2,
                                 int NB, int segLen,
                                 float* __restrict__ bestV, int* __restrict__ bestI) {
  __shared__ float4 sBP[TILE];
  __shared__ float  sA1[TILE];

  int seg = blockIdx.y;
  int j = blockIdx.x * blockDim.x + threadIdx.x;
  bool active = (j < NB);
  float4 t = active ? BT[j] : make_float4(0.f, 0.f, 0.f, 0.f);
  float ta = active ? A2[j] : 0.f;

  int r0 = seg * segLen;
  int r1 = min(r0 + segLen, NB);

  float best = -1.0f;
  int bidx = 0;

  for (int base = r0; base < r1; base += TILE) {
    int i = base + threadIdx.x;
    if (i < r1) { sBP[threadIdx.x] = BP[i]; sA1[threadIdx.x] = A1[i]; }
    __syncthreads();
    int lim = min(TILE, r1 - base);
    if (active) {
#pragma unroll 4
      for (int kk = 0; kk < lim; ++kk) {
        float4 p = sBP[kk];           // broadcast read: conflict-free
        float pa = sA1[kk];
        float xl = fmaxf(p.x, t.x), yl = fmaxf(p.y, t.y);
        float xr = fminf(p.z, t.z), yr = fminf(p.w, t.w);
        float w = fmaxf(xr - xl, 0.f), h = fmaxf(yr - yl, 0.f);
        float inter = w * h;
        float iou = inter / (pa + ta - inter);
        if (iou > best) { best = iou; bidx = base + kk; }  // first-max (ascending)
      }
    }
    __syncthreads();
  }

  if (active) {
    bestV[seg * NB + j] = best;
    bestI[seg * NB + j] = bidx;
  }
}

__global__ void yolo_iou_combine(const float4* __restrict__ BP, const float4* __restrict__ BT,
                                 const float* __restrict__ ConfP,
                                 const float* __restrict__ bestV, const int* __restrict__ bestI,
                                 int NB, float* __restrict__ partial) {
  __shared__ float red[256];
  int j = blockIdx.x * blockDim.x + threadIdx.x;
  bool active = (j < NB);

  float contrib = 0.f;
  if (active) {
    float best = -1.0f;
    int bidx = 0;
    // ascending segment order + strict > == global first-occurrence argmax
#pragma unroll
    for (int seg = 0; seg < NSEG; ++seg) {
      float v = bestV[seg * NB + j];
      if (v > best) { best = v; bidx = bestI[seg * NB + j]; }
    }
    float4 pg = BP[bidx];
    float4 tg = BT[bidx];   // reference indexes BT with argmax row too
    float dx = pg.x - tg.x, dy = pg.y - tg.y;
    float lXY = dx * dx + dy * dy;
    float sw = sqrtf(pg.z) - sqrtf(tg.z);
    float sh = sqrtf(pg.w) - sqrtf(tg.w);
    float lWH = sw * sw + sh * sh;
    float dO = ConfP[bidx] - best;
    contrib = 5.0f * (lXY + lWH) + dO * dO;   // LAMBDA_OBJ*(XY+WH) + Obj
  }

  red[threadIdx.x] = contrib;
  __syncthreads();
#pragma unroll
  for (int s = 128; s > 0; s >>= 1) {
    if ((int)threadIdx.x < s) red[threadIdx.x] += red[threadIdx.x + s];
    __syncthreads();
  }
  if (threadIdx.x == 0) partial[blockIdx.x] = red[0];
}

// Class stream element t (32-bit, branch-free, clamped-index + select):
//   cell = 8*(t/20), col = 10 + t%20, weight 1
__device__ __forceinline__ float class_elem(const float* __restrict__ P,
                                            const float* __restrict__ T,
                                            int t, int lenA) {
  int tc = min(t, lenA - 1);
  int k = tc / 20;                    // 32-bit magic-mul
  int idx = k * (8 * N_CH) + (tc - k * 20) + 10;
  float d = P[idx] - T[idx];
  float v = d * d;
  return (t < lenA) ? v : 0.f;
}

// NoObj stream element u: q = u/2, cell = 8*(q/7) + q%7 + 1, col = u&1 ? 9 : 4, weight 0.5
__device__ __forceinline__ float noobj_elem(const float* __restrict__ P,
                                            const float* __restrict__ T,
                                            int u, int lenB) {
  int uc = min(u, lenB - 1);
  int q = uc >> 1;
  int col = (uc & 1) ? 9 : 4;
  int g = q / 7;                      // 32-bit magic-mul
  int idx = ((g << 3) + (q - g * 7) + 1) * N_CH + col;
  float d = P[idx] - T[idx];
  float v = 0.5f * d * d;             // LAMBDA_NOOBJ, power of two -> exact
  return (u < lenB) ? v : 0.f;
}

__global__ void yolo_sse_wmma(const float* __restrict__ P, const float* __restrict__ T,
                              int CELLS, int OBJ, float* __restrict__ partial) {
  __shared__ float red[256];
  int lane = threadIdx.x & 31;
  // 32-bit A-matrix 16x4 layout: lane holds (m = lane&15, k = 2*(lane>>4)) and k+1.
  // Flatten element index within a 64-elem tile as e = m*4 + k.
  int e0 = ((lane & 15) << 2) + ((lane >> 4) << 1);

  int lenA = OBJ * 20;
  int lenB = (CELLS - OBJ) * 2;
  int ntilesA = (lenA + 63) >> 6;
  int ntilesB = (lenB + 63) >> 6;

  int wavesPerBlock = blockDim.x >> 5;
  int wid = blockIdx.x * wavesPerBlock + (threadIdx.x >> 5);
  int nwaves = gridDim.x * wavesPerBlock;

  v8f c = {0.f, 0.f, 0.f, 0.f, 0.f, 0.f, 0.f, 0.f};
  v2f ones = {1.0f, 1.0f};

  // Class stream (weight 1). Uniform per-wave trip count; EXEC all-ones at WMMA.
  for (int tile = wid; tile < ntilesA; tile += nwaves) {
    int base = tile << 6;
    v2f a = {class_elem(P, T, base + e0, lenA),
             class_elem(P, T, base + e0 + 1, lenA)};
    // D[m,n] = rowsum_m(A) + C[m,n] for all 16 columns (B = ones)
    c = __builtin_amdgcn_wmma_f32_16x16x4_f32(
        false, a, false, ones, (short)0, c, false, false);
  }
  // NoObj stream (weight 0.5, pre-scaled exactly).
  for (int tile = wid; tile < ntilesB; tile += nwaves) {
    int base = tile << 6;
    v2f a = {noobj_elem(P, T, base + e0, lenB),
             noobj_elem(P, T, base + e0 + 1, lenB)};
    c = __builtin_amdgcn_wmma_f32_16x16x4_f32(
        false, a, false, ones, (short)0, c, false, false);
  }

  float s = c[0] + c[1] + c[2] + c[3] + c[4] + c[5] + c[6] + c[7];
  red[threadIdx.x] = s;
  __syncthreads();
#pragma unroll
  for (int st = 128; st > 0; st >>= 1) {
    if ((int)threadIdx.x < st) red[threadIdx.x] += red[threadIdx.x + st];
    __syncthreads();
  }
  // every D column carries the same sum -> divide replicated total by 16
  if (threadIdx.x == 0) partial[blockIdx.x] = red[0] * 0.0625f;
}

__global__ void yolo_final(const float* __restrict__ p2, int n2,
                           const float* __restrict__ p3, int n3,
                           float invBatch, float* __restrict__ out) {
  if (threadIdx.x == 0 && blockIdx.x == 0) {
    float s = 0.f;
    for (int i = 0; i < n2; ++i) s += p2[i];
    for (int i = 0; i < n3; ++i) s += p3[i];
    out[0] = s * invBatch;
  }
}

extern "C" void kernel_launch(void* const* d_in, const int* in_sizes, int n_in,
                              void* d_out, int out_size, void* d_ws, size_t ws_size,
                              hipStream_t stream) {
  const float* P = (const float*)d_in[0];
  const float* T = (const float*)d_in[1];

  int total = in_sizes[0];
  int CELLS = total / N_CH;          // 25088
  int batch = CELLS / 49;            // 512
  int OBJ = (CELLS + 7) / 8;         // 3136
  int NB = 2 * OBJ;                  // 6272

  // workspace layout (16B-aligned chunks first)
  char* ws = (char*)d_ws;
  float4* BP = (float4*)ws;   ws += (size_t)NB * sizeof(float4);
  float4* BT = (float4*)ws;   ws += (size_t)NB * sizeof(float4);
  float* ConfP = (float*)ws;  ws += (size_t)NB * sizeof(float);
  float* A1 = (float*)ws;     ws += (size_t)NB * sizeof(float);
  float* A2 = (float*)ws;     ws += (size_t)NB * sizeof(float);
  float* bestV = (float*)ws;  ws += (size_t)NSEG * NB * sizeof(float);
  int*   bestI = (int*)ws;    ws += (size_t)NSEG * NB * sizeof(int);

  int nblkCols = (NB + 255) / 256;   // 25
  const int NBLK3 = 64;
  float* part2 = (float*)ws;  ws += (size_t)nblkCols * sizeof(float);
  float* part3 = (float*)ws;  ws += (size_t)NBLK3 * sizeof(float);

  int segLen = (NB + NSEG - 1) / NSEG;   // 784

  yolo_prep<<<nblkCols, 256, 0, stream>>>(P, T, NB, BP, BT, ConfP, A1, A2);
  yolo_iou_partial<<<dim3(nblkCols, NSEG), 256, 0, stream>>>(BP, BT, A1, A2, NB, segLen,
                                                             bestV, bestI);
  yolo_iou_combine<<<nblkCols, 256, 0, stream>>>(BP, BT, ConfP, bestV, bestI, NB, part2);
  yolo_sse_wmma<<<NBLK3, 256, 0, stream>>>(P, T, CELLS, OBJ, part3);
  yolo_final<<<1, 1, 0, stream>>>(part2, nblkCols, part3, NBLK3, 1.0f / (float)batch,
                                  (float*)d_out);
}